// VideoHungarianMatcherProjMask_48885317763706
// MI455X (gfx1250) — compile-verified
//
#include <hip/hip_runtime.h>
#include <hip/hip_bf16.h>
#include <math.h>

// Problem dims (fixed by setup_inputs)
#define B_ 2
#define Q_ 100
#define G_ 10
#define T_ 5
#define H_ 128
#define W_ 128
#define C_ 41
#define KDIM (T_*H_)   // 640, also T_*W_

typedef __attribute__((ext_vector_type(2))) float v2f;
typedef __attribute__((ext_vector_type(8))) float v8f;

// ---------------- workspace layout (bytes) ----------------
#define OFF_BOUNDS 0                         // int [B][G][T][4]  (y0,y1,x0,x1)
#define OFF_PROJY  2048                      // float [B][Q][T*H]
#define SZ_PROJ    (B_*Q_*KDIM*4)            // 512000
#define OFF_PROJX  (OFF_PROJY + SZ_PROJ)
#define OFF_SFGY   (OFF_PROJX + SZ_PROJ)     // float [B][Q][G][T]
#define SZ_SFG     (B_*Q_*G_*T_*4)
#define OFF_SFGX   (OFF_SFGY + SZ_SFG)
#define OFF_SFULL  (OFF_SFGX + SZ_SFG)       // float [B][Q][T][2] (y,x)
#define SZ_SFULL   (B_*Q_*T_*2*4)
#define OFF_INBOX  (OFF_SFULL + SZ_SFULL)    // float [B][2][Q][G]

__device__ __forceinline__ float sigf(float x) { return 1.0f / (1.0f + expf(-x)); }

// ---------------- kernel 1: box bounds ----------------
// grid = B*G*T blocks of 128 threads; each block scans one (b,g,t) mask.
__global__ void k_bounds(const float* __restrict__ box, int* __restrict__ bnds) {
  const int bid = blockIdx.x;                 // (b*G+g)*T + t
  const float* m = box + (size_t)bid * (H_ * W_);
  __shared__ int rowAny[H_];
  __shared__ int colAny[W_];
  const int tid = threadIdx.x;                // 0..127

  {   // row tid
    int any = 0;
    const float* r = m + tid * W_;
    for (int x = 0; x < W_; ++x) any |= (r[x] > 0.5f) ? 1 : 0;
    rowAny[tid] = any;
  }
  {   // col tid (coalesced across lanes)
    int any = 0;
    for (int y = 0; y < H_; ++y) any |= (m[y * W_ + tid] > 0.5f) ? 1 : 0;
    colAny[tid] = any;
  }
  __syncthreads();
  if (tid == 0) {
    int y0 = -1, y1 = -1;
    for (int y = 0; y < H_; ++y) if (rowAny[y]) { if (y0 < 0) y0 = y; y1 = y + 1; }
    if (y0 < 0) { y0 = 0; y1 = 0; }           // empty mask -> (0,0) like reference
    bnds[bid * 4 + 0] = y0; bnds[bid * 4 + 1] = y1;
  } else if (tid == 1) {
    int x0 = -1, x1 = -1;
    for (int x = 0; x < W_; ++x) if (colAny[x]) { if (x0 < 0) x0 = x; x1 = x + 1; }
    if (x0 < 0) { x0 = 0; x1 = 0; }
    bnds[bid * 4 + 2] = x0; bnds[bid * 4 + 3] = x1;
  }
}

// ---------------- kernel 2: streaming projections ----------------
// grid = B*Q*T blocks of 256 threads. One 128x128 sigmoid tile in LDS
// (row pitch 129 floats -> conflict-free row and column walks).
__global__ __launch_bounds__(256) void k_main(
    const float* __restrict__ pm, const int* __restrict__ bnds,
    float* __restrict__ projy, float* __restrict__ projx,
    float* __restrict__ sfgy, float* __restrict__ sfgx,
    float* __restrict__ sfull) {
  const int bid = blockIdx.x;                 // (b*Q+q)*T + t
  const int t  = bid % T_;
  const int bq = bid / T_;                    // b*Q + q
  const int b  = bq / Q_;
  const int tid = threadIdx.x;

  __shared__ float tile[H_ * (W_ + 1)];
  __shared__ float redA[128];
  __shared__ float redB[128];

  // ---- load tile, apply sigmoid ----
  const float4* src = (const float4*)(pm + (size_t)bid * (H_ * W_));
  #pragma unroll
  for (int c = 0; c < 16; ++c) {
    int idx4 = c * 256 + tid;                 // 0..4095 float4s
    float4 v = src[idx4];
    int e = idx4 * 4;
    int y = e >> 7, x = e & 127;
    float* d = &tile[y * (W_ + 1) + x];
    d[0] = sigf(v.x); d[1] = sigf(v.y); d[2] = sigf(v.z); d[3] = sigf(v.w);
  }
  __syncthreads();

  // ---- full row/col maxima (4-way ILP on the reduction chains) ----
  if (tid < 128) {
    const int y = tid;
    const float* r = &tile[y * (W_ + 1)];
    float m0 = r[0], m1 = r[1], m2 = r[2], m3 = r[3];
    for (int x = 4; x < W_; x += 4) {
      m0 = fmaxf(m0, r[x + 0]); m1 = fmaxf(m1, r[x + 1]);
      m2 = fmaxf(m2, r[x + 2]); m3 = fmaxf(m3, r[x + 3]);
    }
    float mx = fmaxf(fmaxf(m0, m1), fmaxf(m2, m3));
    projy[((size_t)bq * T_ + t) * H_ + y] = mx;
    redA[y] = mx;
  } else {
    const int x = tid - 128;
    float m0 = tile[x], m1 = tile[(W_ + 1) + x];
    float m2 = tile[2 * (W_ + 1) + x], m3 = tile[3 * (W_ + 1) + x];
    for (int y = 4; y < H_; y += 4) {
      m0 = fmaxf(m0, tile[(y + 0) * (W_ + 1) + x]);
      m1 = fmaxf(m1, tile[(y + 1) * (W_ + 1) + x]);
      m2 = fmaxf(m2, tile[(y + 2) * (W_ + 1) + x]);
      m3 = fmaxf(m3, tile[(y + 3) * (W_ + 1) + x]);
    }
    float mx = fmaxf(fmaxf(m0, m1), fmaxf(m2, m3));
    projx[((size_t)bq * T_ + t) * W_ + x] = mx;
    redB[x] = mx;
  }
  __syncthreads();
  for (int s = 64; s > 0; s >>= 1) {
    if (tid < s) redA[tid] += redA[tid + s];
    else if (tid >= 128 && tid < 128 + s) redB[tid - 128] += redB[tid - 128 + s];
    __syncthreads();
  }
  if (tid == 0)   sfull[((size_t)bq * T_ + t) * 2 + 0] = redA[0];
  if (tid == 128) sfull[((size_t)bq * T_ + t) * 2 + 1] = redB[0];

  // ---- per-target box-restricted maxima ----
  for (int g = 0; g < G_; ++g) {
    const int* bb = &bnds[((b * G_ + g) * T_ + t) * 4];
    const int y0 = bb[0], y1 = bb[1], x0 = bb[2], x1 = bb[3];
    __syncthreads();                          // protect redA[0]/redB[0] readers
    float acc = 0.0f;
    if (tid < 128) {                          // row maxima inside box
      for (int y = y0 + tid; y < y1; y += 128) {
        float mx = 0.0f;
        for (int x = x0; x < x1; ++x) mx = fmaxf(mx, tile[y * (W_ + 1) + x]);
        acc += mx;
      }
      redA[tid] = acc;
    } else {                                  // col maxima inside box
      for (int x = x0 + (tid - 128); x < x1; x += 128) {
        float mx = 0.0f;
        for (int y = y0; y < y1; ++y) mx = fmaxf(mx, tile[y * (W_ + 1) + x]);
        acc += mx;
      }
      redB[tid - 128] = acc;
    }
    __syncthreads();
    for (int s = 64; s > 0; s >>= 1) {
      if (tid < s) redA[tid] += redA[tid + s];
      else if (tid >= 128 && tid < 128 + s) redB[tid - 128] += redB[tid - 128 + s];
      __syncthreads();
    }
    if (tid == 0)   sfgy[((size_t)bq * G_ + g) * T_ + t] = redA[0];
    if (tid == 128) sfgx[((size_t)bq * G_ + g) * T_ + t] = redB[0];
  }
}

// ---------------- kernel 3: WMMA indicator GEMM ----------------
// Inbox[b,axis,q,g] = sum_k proj_full[b,q,k] * ind[b,axis,k,g]
// via V_WMMA_F32_16X16X4_F32; one wave per 16x16 output tile.
// grid = B * 2(axes) * 7(M-tiles) blocks of 32 threads.
// Inner loop is branchless: frame index t is uniform per 32 k-steps, so the
// box bounds are hoisted and the indicator is pure VALU (cmp + cndmask).
__global__ void k_inbox(const float* __restrict__ projy, const float* __restrict__ projx,
                        const int* __restrict__ bnds, float* __restrict__ inbox) {
  const int bid   = blockIdx.x;
  const int b     = bid / 14;
  const int rem   = bid % 14;
  const int axis  = rem / 7;                  // 0=y, 1=x
  const int mtile = rem % 7;
  const int q0    = mtile * 16;
  const int lane  = threadIdx.x;              // 0..31, wave32

  __shared__ int lo[G_ * T_];
  __shared__ int hi[G_ * T_];
  for (int i = lane; i < G_ * T_; i += 32) {
    const int* bb = &bnds[(b * G_ * T_ + i) * 4];
    lo[i] = bb[axis * 2 + 0];
    hi[i] = bb[axis * 2 + 1];
  }
  __syncthreads();

  const float* P = (axis == 0 ? projy : projx) + (size_t)b * Q_ * KDIM;

  const int m  = lane & 15;                   // A row (query within tile)
  const int h  = lane >> 4;                   // K half (0 or 1)
  int qa = q0 + m; if (qa > Q_ - 1) qa = Q_ - 1;   // clamp padded rows
  const int n  = lane & 15;                   // B col (target)
  const int gi = (n < G_) ? n : 0;
  const float gmask = (n < G_) ? 1.0f : 0.0f;
  const float* Arow = P + (size_t)qa * KDIM;

  v8f c = {};
  for (int t = 0; t < T_; ++t) {              // frame-major: bounds hoisted
    const int lo_t = lo[gi * T_ + t];
    const int hi_t = hi[gi * T_ + t];
    const float* At = Arow + t * H_;
    for (int kk2 = 0; kk2 < H_ / 4; ++kk2) {  // 32 K-steps of 4 per frame
      const int r0 = kk2 * 4 + h * 2;         // row index within frame
      const int r1 = r0 + 1;
      v2f a, bf;
      a.x = At[r0];
      a.y = At[r1];
      bf.x = (((r0 >= lo_t) & (r0 < hi_t)) != 0) ? gmask : 0.0f;
      bf.y = (((r1 >= lo_t) & (r1 < hi_t)) != 0) ? gmask : 0.0f;
      // D = A(16x4) * B(4x16) + C   (f32, accumulate over K loop)
      c = __builtin_amdgcn_wmma_f32_16x16x4_f32(
          /*neg_a=*/false, a, /*neg_b=*/false, bf,
          /*c_mod=*/(short)0, c, /*reuse_a=*/false, /*reuse_b=*/false);
    }
  }

  #pragma unroll
  for (int j = 0; j < 8; ++j) {               // D: vgpr j -> M = j + h*8, N = lane&15
    int q = q0 + j + h * 8;
    int g = lane & 15;
    if (q < Q_ && g < G_)
      inbox[(((size_t)b * 2 + axis) * Q_ + q) * G_ + g] = c[j];
  }
}

// ---------------- kernel 4: combine ----------------
__global__ void k_final(const float* __restrict__ logits, const int* __restrict__ tgt,
                        const int* __restrict__ bnds,
                        const float* __restrict__ sfgy, const float* __restrict__ sfgx,
                        const float* __restrict__ sfull, const float* __restrict__ inbox,
                        float* __restrict__ out) {
  const int idx = blockIdx.x * blockDim.x + threadIdx.x;
  if (idx >= B_ * Q_ * G_) return;
  const int g  = idx % G_;
  const int bq = idx / G_;                    // b*Q + q
  const int q  = bq % Q_;
  const int b  = bq / Q_;

  // -softmax(logits)[q, tgt_id]
  const float* L = logits + (size_t)bq * C_;
  float mx = L[0];
  for (int i = 1; i < C_; ++i) mx = fmaxf(mx, L[i]);
  float s = 0.0f;
  for (int i = 0; i < C_; ++i) s += expf(L[i] - mx);
  const int cls = tgt[b * G_ + g];
  const float cost_class = -(expf(L[cls] - mx) / s);

  float Sfy = 0.f, Sfx = 0.f, Fully = 0.f, Fullx = 0.f, Ay = 0.f, Ax = 0.f;
  for (int t = 0; t < T_; ++t) {
    Sfy   += sfgy[((size_t)bq * G_ + g) * T_ + t];
    Sfx   += sfgx[((size_t)bq * G_ + g) * T_ + t];
    Fully += sfull[((size_t)bq * T_ + t) * 2 + 0];
    Fullx += sfull[((size_t)bq * T_ + t) * 2 + 1];
    const int* bb = &bnds[((b * G_ + g) * T_ + t) * 4];
    Ay += (float)(bb[1] - bb[0]);
    Ax += (float)(bb[3] - bb[2]);
  }
  const float Iy = inbox[(((size_t)b * 2 + 0) * Q_ + q) * G_ + g];
  const float Ix = inbox[(((size_t)b * 2 + 1) * Q_ + q) * G_ + g];

  const float dice_y = 1.0f - (2.0f * Sfy + 1.0f) / (Sfy + (Fully - Iy) + Ay + 1.0f);
  const float dice_x = 1.0f - (2.0f * Sfx + 1.0f) / (Sfx + (Fullx - Ix) + Ax + 1.0f);

  out[idx] = 2.0f * cost_class + 5.0f * (dice_y + dice_x);
}

extern "C" void kernel_launch(void* const* d_in, const int* in_sizes, int n_in,
                              void* d_out, int out_size, void* d_ws, size_t ws_size,
                              hipStream_t stream) {
  (void)in_sizes; (void)n_in; (void)out_size; (void)ws_size;
  const float* logits = (const float*)d_in[0];   // (B,Q,C)
  const float* pmask  = (const float*)d_in[1];   // (B,Q,T,H,W)
  const float* bmask  = (const float*)d_in[2];   // (B,G,T,H,W)
  const int*   tgt    = (const int*)d_in[3];     // (B,G)
  float* out = (float*)d_out;                    // (B,Q,G)
  char*  ws  = (char*)d_ws;

  int*   bnds  = (int*)(ws + OFF_BOUNDS);
  float* projy = (float*)(ws + OFF_PROJY);
  float* projx = (float*)(ws + OFF_PROJX);
  float* sfgy  = (float*)(ws + OFF_SFGY);
  float* sfgx  = (float*)(ws + OFF_SFGX);
  float* sfull = (float*)(ws + OFF_SFULL);
  float* inbox = (float*)(ws + OFF_INBOX);

  k_bounds<<<B_ * G_ * T_, 128, 0, stream>>>(bmask, bnds);
  k_main  <<<B_ * Q_ * T_, 256, 0, stream>>>(pmask, bnds, projy, projx, sfgy, sfgx, sfull);
  k_inbox <<<B_ * 2 * 7, 32, 0, stream>>>(projy, projx, bnds, inbox);
  k_final <<<(B_ * Q_ * G_ + 127) / 128, 128, 0, stream>>>(logits, tgt, bnds,
                                                           sfgy, sfgx, sfull, inbox, out);
}